// KernelRepeatLinear_12481174962958
// MI455X (gfx1250) — compile-verified
//
#include <hip/hip_runtime.h>
#include <stdint.h>

typedef __attribute__((ext_vector_type(2))) float v2f;
typedef __attribute__((ext_vector_type(8))) float v8f;

#define BDIM   8
#define EDIM   2048
#define SDIM   512
#define KTAPS  8
#define HALO   (KTAPS - 1)
#define TILE_E 24
#define ROWS   (TILE_E + HALO)      // 31 rows
#define STRIDE 516                  // padded row stride (floats): 4-word bank skew, 16B aligned
#define NTHREADS 256
#define NWAVES   8
#define ROWS_PER_WAVE (TILE_E / NWAVES)   // 3
#define NBLK   (SDIM / 16)          // 32 feature blocks of 16
#define NGRP   2                    // row groups: rows 0-15, rows 16-30 (clamped)
#define NTASK  (NGRP * NBLK)        // 64 independent WMMA tasks per block

#define HAVE_WMMA_F32X4 __has_builtin(__builtin_amdgcn_wmma_f32_16x16x4_f32)

// CDNA5 async global->LDS copy (tracked by ASYNCcnt).
__device__ __forceinline__ void async_copy_b128(uint32_t lds_byte_addr, const void* gaddr) {
    asm volatile("global_load_async_to_lds_b128 %0, %1, off"
                 :: "v"(lds_byte_addr), "v"((uint64_t)(uintptr_t)gaddr)
                 : "memory");
}
__device__ __forceinline__ void wait_async_all() {
    asm volatile("s_wait_asynccnt 0" ::: "memory");
}

__global__ __launch_bounds__(NTHREADS)
void krl_scan_conv_kernel(const float* __restrict__ x,
                          const float* __restrict__ w,
                          const float* __restrict__ bias,
                          const float* __restrict__ decay,
                          float* __restrict__ out) {
    __shared__ float tile[ROWS * STRIDE];   // 63,984 bytes

    const int tid  = (int)threadIdx.x;
    const int wave = tid >> 5;
    const int lane = tid & 31;
    const int e0   = (int)blockIdx.x * TILE_E;
    const int b    = (int)blockIdx.y;

    // dv = clip(decay_value[1,0], 0.9, 1.0); DECAY_CONSTANT == 1 -> per-element ratio r1 = dv
    float dv = decay[1];
    dv = fminf(fmaxf(dv, 0.9f), 1.0f);
    const float r1  = dv;
    const float r2  = r1 * r1;
    const float r4  = r2 * r2;
    const float r8  = r4 * r4;
    const float r16 = r8 * r8;               // per-16-feature-block decay

    // ---------------- Phase 1a: stage x rows into LDS via async DMA ----------------
    const uint32_t lds_base = (uint32_t)(uintptr_t)(&tile[0]);
    for (int m = wave; m < ROWS; m += NWAVES) {
        const int e_row = e0 - HALO + m;
        if (e_row >= 0 && e_row < EDIM) {
            const char* grow = (const char*)(x + ((size_t)b * EDIM + e_row) * SDIM);
            const uint32_t lrow = lds_base + (uint32_t)(m * STRIDE * 4);
            #pragma unroll
            for (int it = 0; it < 4; ++it) {
                const uint32_t off = (uint32_t)(it * 512 + lane * 16);
                async_copy_b128(lrow + off, grow + off);
            }
        } else {
            float4 z = make_float4(0.f, 0.f, 0.f, 0.f);
            float4* lrow4 = (float4*)&tile[m * STRIDE];
            #pragma unroll
            for (int it = 0; it < 4; ++it)
                lrow4[it * 32 + lane] = z;
        }
    }
    wait_async_all();        // each wave drains its own ASYNCcnt, then cross-wave barrier
    __syncthreads();

#if HAVE_WMMA_F32X4
    // ------- Phase 1b (Pass A): intra-block triangular scan on the matrix pipe -----
    // P(16 rows x 16 feats) = X_blk @ T,  T[i][j] = r1^(j-i) for j>=i else 0,
    // via 4 chained V_WMMA_F32_16X16X4_F32 (K-chunks of 4). In-place: x -> c-partial.
    {
        const int n    = lane & 15;            // B/C/D column owned by this lane
        const int koff = (lane >> 4) << 1;     // A/B: hi half-wave carries K+2
        const int m16  = lane & 15;            // A: lanes 0-15 hold rows M=0-15
        const int mbase = (lane >> 4) << 3;    // D: hi half-wave holds M=8-15

        // B chunks (constant per wave): B[kc] holds T rows {4kc+koff, 4kc+koff+1}
        v2f Bc[4];
        #pragma unroll
        for (int kc = 0; kc < 4; ++kc) {
            const int i0 = 4 * kc + koff;
            Bc[kc][0] = (n >= i0    ) ? __powf(r1, (float)(n - i0    )) : 0.f;
            Bc[kc][1] = (n >= i0 + 1) ? __powf(r1, (float)(n - i0 - 1)) : 0.f;
        }

        for (int t = wave; t < NTASK; t += NWAVES) {
            const int grp = t >> 5;            // row group
            const int g   = t & (NBLK - 1);    // feature block
            int arow = grp * 16 + m16;
            if (arow > ROWS - 1) arow = ROWS - 1;          // clamp pad row (group 1)
            const float* abase = &tile[arow * STRIDE + g * 16 + koff];

            v8f acc = {};
            #pragma unroll
            for (int kc = 0; kc < 4; ++kc) {
                const float2 af = *(const float2*)(abase + 4 * kc);  // ds_load_b64, skewed banks
                v2f a; a[0] = af.x; a[1] = af.y;
                acc = __builtin_amdgcn_wmma_f32_16x16x4_f32(
                          false, a, false, Bc[kc], (short)0, acc, false, false);
            }
            #pragma unroll
            for (int v = 0; v < 8; ++v) {
                const int drow = grp * 16 + mbase + v;
                if (drow < ROWS)
                    tile[drow * STRIDE + g * 16 + n] = acc[v];
            }
        }
    }
    __syncthreads();

    // ------- Phase 1b (Pass B): cross-block carry scan + rank-1 fixup ---------------
    {
        // fixup multipliers r1^(i+1), i = 0..15, packed as 4 float4 (static chain)
        float e1 = r1;
        float4 M0, M1, M2, M3;
        M0.x = e1; e1 *= r1; M0.y = e1; e1 *= r1; M0.z = e1; e1 *= r1; M0.w = e1; e1 *= r1;
        M1.x = e1; e1 *= r1; M1.y = e1; e1 *= r1; M1.z = e1; e1 *= r1; M1.w = e1; e1 *= r1;
        M2.x = e1; e1 *= r1; M2.y = e1; e1 *= r1; M2.z = e1; e1 *= r1; M2.w = e1; e1 *= r1;
        M3.x = e1; e1 *= r1; M3.y = e1; e1 *= r1; M3.z = e1; e1 *= r1; M3.w = e1;

        for (int m = wave; m < ROWS; m += NWAVES) {
            // lane g holds this block's decayed sum = c-partial at feature 16g+15
            float t = tile[m * STRIDE + lane * 16 + 15];
            float f = r16;
            #pragma unroll
            for (int d = 1; d < 32; d <<= 1) {       // Kogge-Stone, weight r16^d
                float u = __shfl_up(t, d, 32);
                if (lane >= d) t = fmaf(u, f, t);
                f = f * f;
            }
            float carry = __shfl_up(t, 1, 32);       // carry entering block g
            if (lane == 0) carry = 0.f;

            float4* cp = (float4*)&tile[m * STRIDE + lane * 16];
            float4 c0 = cp[0], c1 = cp[1], c2 = cp[2], c3 = cp[3];
            c0.x = fmaf(carry, M0.x, c0.x); c0.y = fmaf(carry, M0.y, c0.y);
            c0.z = fmaf(carry, M0.z, c0.z); c0.w = fmaf(carry, M0.w, c0.w);
            c1.x = fmaf(carry, M1.x, c1.x); c1.y = fmaf(carry, M1.y, c1.y);
            c1.z = fmaf(carry, M1.z, c1.z); c1.w = fmaf(carry, M1.w, c1.w);
            c2.x = fmaf(carry, M2.x, c2.x); c2.y = fmaf(carry, M2.y, c2.y);
            c2.z = fmaf(carry, M2.z, c2.z); c2.w = fmaf(carry, M2.w, c2.w);
            c3.x = fmaf(carry, M3.x, c3.x); c3.y = fmaf(carry, M3.y, c3.y);
            c3.z = fmaf(carry, M3.z, c3.z); c3.w = fmaf(carry, M3.w, c3.w);
            cp[0] = c0; cp[1] = c1; cp[2] = c2; cp[3] = c3;
        }
    }
#else
    // ------- Fallback: per-row VALU scan (lane owns 16 contiguous features) ---------
    for (int m = wave; m < ROWS; m += NWAVES) {
        float* lrow = &tile[m * STRIDE];
        float v[16];
        #pragma unroll
        for (int i = 0; i < 4; ++i) {
            float4 a = ((float4*)lrow)[lane * 4 + i];
            v[4*i+0] = a.x; v[4*i+1] = a.y; v[4*i+2] = a.z; v[4*i+3] = a.w;
        }
        float s = 0.f;
        #pragma unroll
        for (int i = 0; i < 16; ++i) { s = fmaf(s, r1, v[i]); v[i] = s; }
        float t = s, f = r16;
        #pragma unroll
        for (int d = 1; d < 32; d <<= 1) {
            float u = __shfl_up(t, d, 32);
            if (lane >= d) t = fmaf(u, f, t);
            f = f * f;
        }
        float carry = __shfl_up(t, 1, 32);
        if (lane == 0) carry = 0.f;
        float mlt = r1;
        #pragma unroll
        for (int i = 0; i < 16; ++i) { v[i] = fmaf(carry, mlt, v[i]); mlt *= r1; }
        #pragma unroll
        for (int i = 0; i < 4; ++i) {
            float4 a;
            a.x = v[4*i+0]; a.y = v[4*i+1]; a.z = v[4*i+2]; a.w = v[4*i+3];
            ((float4*)lrow)[lane * 4 + i] = a;
        }
    }
#endif
    __syncthreads();

    // ---------------- Phase 2: 8-tap causal depthwise conv along E + bias ----------
    // out[e, j] = bias[j] + sum_k weight[k, j] * c[e + k - 7, j]
    const int rbase = wave * ROWS_PER_WAVE;
    #pragma unroll
    for (int i = 0; i < 4; ++i) {
        const int j0 = i * 128 + lane * 4;          // contiguous across lanes
        const float4 bv = *(const float4*)(bias + j0);
        float4 acc[ROWS_PER_WAVE];
        #pragma unroll
        for (int rr = 0; rr < ROWS_PER_WAVE; ++rr) acc[rr] = bv;
        #pragma unroll
        for (int k = 0; k < KTAPS; ++k) {
            const float4 wv = *(const float4*)(w + k * SDIM + j0);
            #pragma unroll
            for (int rr = 0; rr < ROWS_PER_WAVE; ++rr) {
                const float4 cv = *(const float4*)&tile[(rbase + rr + k) * STRIDE + j0];
                acc[rr].x = fmaf(wv.x, cv.x, acc[rr].x);
                acc[rr].y = fmaf(wv.y, cv.y, acc[rr].y);
                acc[rr].z = fmaf(wv.z, cv.z, acc[rr].z);
                acc[rr].w = fmaf(wv.w, cv.w, acc[rr].w);
            }
        }
        #pragma unroll
        for (int rr = 0; rr < ROWS_PER_WAVE; ++rr) {
            const int e = e0 + rbase + rr;
            if (e < EDIM)
                *(float4*)(out + ((size_t)b * EDIM + e) * SDIM + j0) = acc[rr];
        }
    }
}

extern "C" void kernel_launch(void* const* d_in, const int* in_sizes, int n_in,
                              void* d_out, int out_size, void* d_ws, size_t ws_size,
                              hipStream_t stream) {
    (void)in_sizes; (void)n_in; (void)out_size; (void)d_ws; (void)ws_size;
    const float* x     = (const float*)d_in[0];
    const float* w     = (const float*)d_in[1];
    const float* bias  = (const float*)d_in[2];
    const float* decay = (const float*)d_in[3];
    float* out = (float*)d_out;

    dim3 grid((EDIM + TILE_E - 1) / TILE_E, BDIM);   // (86, 8)
    krl_scan_conv_kernel<<<grid, NTHREADS, 0, stream>>>(x, w, bias, decay, out);
}